// SFTBranch_12584254177491
// MI455X (gfx1250) — compile-verified
//
#include <hip/hip_runtime.h>
#include <math.h>

// Problem constants (from reference)
#define BATCH 2
#define CIN   256
#define COUTC 256
#define Hdim  96
#define Wdim  96
#define KS    3
#define PADC  1
#define GRP   4
#define KKC   9             // KS*KS
#define HWC   (Hdim*Wdim)   // 9216
#define KDIM  (CIN*KKC)     // 2304

// Tiling
#define M_TILE 64
#define NSTEPS 72           // 9 taps * 8 channel-chunks of 32
#define GK     36           // GRP*KKC sampling tuples per pixel

typedef __attribute__((ext_vector_type(2))) float v2f;
typedef __attribute__((ext_vector_type(8))) float v8f;

// ---------------------------------------------------------------------------
// Weight transpose: w_deform (COUT, CIN*9, c-minor k) -> wre k-major layout
//   wre[((k*64 + cq)*256 + o)*4 + cr] = w[o][ (cq*4+cr)*9 + k ]
// so a WMMA B fragment (K-quad = 4 consecutive c at fixed tap k) is one
// contiguous, wave-coalesced 8-byte load.
// ---------------------------------------------------------------------------
__global__ __launch_bounds__(256) void transpose_weights(
    const float* __restrict__ w, float* __restrict__ wre)
{
    int dst = blockIdx.x * 256 + threadIdx.x;   // 0 .. 589823 (= 9<<16)
    int cr = dst & 3;
    int o  = (dst >> 2) & 255;
    int cq = (dst >> 10) & 63;
    int k  = dst >> 16;
    wre[dst] = w[(size_t)o * KDIM + (cq * 4 + cr) * KKC + k];
}

// ---------------------------------------------------------------------------
// Fast path: k-major K order, scratch-transposed weights.
// ---------------------------------------------------------------------------
__global__ __launch_bounds__(256) void deform_conv_wmma_fast(
    const float* __restrict__ feature,    // (B, CIN, H, W)
    const float* __restrict__ pred_shape, // (B, H*W, 4)
    const float* __restrict__ w_offset,   // (72, 4, 1, 1)
    const float* __restrict__ wre,        // transposed weights (2.36 MB)
    float* __restrict__ out)              // (B, COUT, H, W)
{
    // Bilinear tables, [gk][pixel] so each step's (uniform) gk is a scalar offset
    __shared__ float wtab[GK * M_TILE * 4]; // float4: validity-folded weights
    __shared__ int   itab[GK * M_TILE * 2]; // int2:   4 clamped indices packed u16

    const int t    = threadIdx.x;
    const int lane = t & 31;
    const int wave = t >> 5;
    const int mw   = wave & 3;   // 16-row M subtile (0..3)
    const int nw   = wave >> 2;  // 128-col N half   (0..1)
    const int half = lane >> 4;  // lanes 0-15: K=(0,1) of quad; 16-31: K=(2,3)
    const int r    = lane & 15;

    const int mBase  = blockIdx.x * M_TILE;   // 288 blocks, never straddle batch
    const int bIdx   = mBase / HWC;
    const int hwBase = mBase - bIdx * HWC;
    const float* fb  = feature + (size_t)bIdx * CIN * HWC;

    // ---- Phase 0: offsets (4->72 1x1 conv) + bilinear precompute (once) ----
    for (int i = 0; i < 9; ++i) {
        int e  = t + i * 256;          // 0..2303
        int p  = e / GK;
        int gk = e - p * GK;
        int g  = gk / KKC;
        int k  = gk - g * KKC;
        int hwp = hwBase + p;
        int yy  = hwp / Wdim;
        int xx  = hwp - yy * Wdim;

        const float* pr = pred_shape + ((size_t)bIdx * HWC + hwp) * 4;
        float p0v = pr[0], p1v = pr[1], p2v = pr[2], p3v = pr[3];
        const float* wo = w_offset + gk * 8;
        float offY = p0v*wo[0] + p1v*wo[1] + p2v*wo[2] + p3v*wo[3];
        float offX = p0v*wo[4] + p1v*wo[5] + p2v*wo[6] + p3v*wo[7];

        int ky = k / KS, kx = k - ky * KS;
        float py = (float)(yy - PADC + ky) + offY;
        float px = (float)(xx - PADC + kx) + offX;
        float y0f = floorf(py), x0f = floorf(px);
        int   y0  = (int)y0f,  x0  = (int)x0f;
        float ly  = py - y0f,  lx  = px - x0f;

        bool yin0 = (unsigned)y0       < (unsigned)Hdim;
        bool yin1 = (unsigned)(y0 + 1) < (unsigned)Hdim;
        bool xin0 = (unsigned)x0       < (unsigned)Wdim;
        bool xin1 = (unsigned)(x0 + 1) < (unsigned)Wdim;

        float4 wv;
        wv.x = (yin0 && xin0) ? (1.f - ly) * (1.f - lx) : 0.f;
        wv.y = (yin0 && xin1) ? (1.f - ly) * lx         : 0.f;
        wv.z = (yin1 && xin0) ? ly * (1.f - lx)         : 0.f;
        wv.w = (yin1 && xin1) ? ly * lx                 : 0.f;

        int cy0 = min(max(y0, 0), Hdim - 1), cy1 = min(max(y0 + 1, 0), Hdim - 1);
        int cx0 = min(max(x0, 0), Wdim - 1), cx1 = min(max(x0 + 1, 0), Wdim - 1);
        int2 ip;
        ip.x = (cy0 * Wdim + cx0) | ((cy0 * Wdim + cx1) << 16);
        ip.y = (cy1 * Wdim + cx0) | ((cy1 * Wdim + cx1) << 16);

        int idx = gk * M_TILE + p;
        *(float4*)&wtab[idx * 4] = wv;
        *(int2*)&itab[idx * 2]   = ip;
    }
    __syncthreads();   // the ONLY barrier; main loop is barrier-free

    v8f acc[8];
    #pragma unroll
    for (int nt = 0; nt < 8; ++nt) acc[nt] = (v8f)0.0f;

    const int p    = mw * 16 + r;                    // this lane's pixel row
    const int voffB = nw * 512 + r * 4 + half * 2;   // per-lane B offset (floats)

    for (int ws = 0; ws < NSTEPS; ++ws) {
        const int k  = ws >> 3;          // tap 0..8
        const int cc = ws & 7;           // channel chunk 0..7
        const int c0 = cc * 32;
        const int gk = ((c0 >> 6) * KKC) + k;   // UNIFORM for whole step

        // Prefetch next step's contiguous 32KB weight block (L2 -> WGP$)
        if (ws + 1 < NSTEPS) {
            const float* nx = wre + (size_t)(ws + 1) * 8192;
            #pragma unroll
            for (int j = 0; j < 4; ++j)
                __builtin_prefetch(nx + (j * 32 + lane) * 64, 0, 3);
        }

        // One table read per lane per step; 4 gather bases built once.
        int idx = gk * M_TILE + p;
        float4 wv = *(const float4*)&wtab[idx * 4];
        int2   ip = *(const int2*)&itab[idx * 2];
        const float* base = fb + (size_t)(c0 + half * 2) * HWC;
        const float* b00 = base + (ip.x & 0xFFFF);
        const float* b01 = base + ((unsigned)ip.x >> 16);
        const float* b10 = base + (ip.y & 0xFFFF);
        const float* b11 = base + ((unsigned)ip.y >> 16);

        // A fragments: all 64 gathers use compile-time immediate offsets.
        v2f af[8];
        #pragma unroll
        for (int kq = 0; kq < 8; ++kq) {
            float vres[2];
            #pragma unroll
            for (int e2 = 0; e2 < 2; ++e2) {
                const int co = (4 * kq + e2) * HWC;   // constant element offset
                float v =        wv.x * b00[co];
                v = fmaf(wv.y, b01[co], v);
                v = fmaf(wv.z, b10[co], v);
                v = fmaf(wv.w, b11[co], v);
                vres[e2] = v;
            }
            af[kq].x = vres[0];
            af[kq].y = vres[1];
        }

        // B fragments: one base + immediates, each load coalesced across the wave
        const float* wb = wre + (size_t)ws * 8192;    // (k*64 + cc*8) * 1024
        #pragma unroll
        for (int kq = 0; kq < 8; ++kq) {
            #pragma unroll
            for (int nt = 0; nt < 8; ++nt) {
                float2 b2 = *(const float2*)(wb + voffB + kq * 1024 + nt * 64);
                v2f bf; bf.x = b2.x; bf.y = b2.y;
                acc[nt] = __builtin_amdgcn_wmma_f32_16x16x4_f32(
                    false, af[kq], false, bf, (short)0, acc[nt], false, false);
            }
        }
    }

    // ---- epilogue: ReLU + vectorized b128 stores ----
    #pragma unroll
    for (int nt = 0; nt < 8; ++nt) {
        int o = nw * 128 + nt * 16 + r;
        float* op = out + ((size_t)bIdx * COUTC + o) * HWC + hwBase + mw * 16 + half * 8;
        float4 lo, hi;
        lo.x = fmaxf(acc[nt][0], 0.f); lo.y = fmaxf(acc[nt][1], 0.f);
        lo.z = fmaxf(acc[nt][2], 0.f); lo.w = fmaxf(acc[nt][3], 0.f);
        hi.x = fmaxf(acc[nt][4], 0.f); hi.y = fmaxf(acc[nt][5], 0.f);
        hi.z = fmaxf(acc[nt][6], 0.f); hi.w = fmaxf(acc[nt][7], 0.f);
        *(float4*)op       = lo;
        *(float4*)(op + 4) = hi;
    }
}

// ---------------------------------------------------------------------------
// Fallback (no scratch): round-2 kernel, ck-consecutive K order, B direct.
// ---------------------------------------------------------------------------
#define WROW 148
#define IROW 74
__global__ __launch_bounds__(256) void deform_conv_wmma_fallback(
    const float* __restrict__ feature, const float* __restrict__ pred_shape,
    const float* __restrict__ w_offset, const float* __restrict__ w_deform,
    float* __restrict__ out)
{
    __shared__ float wtab[M_TILE * WROW];
    __shared__ int   itab[M_TILE * IROW];
    const int t = threadIdx.x, lane = t & 31, wave = t >> 5;
    const int mw = wave & 3, nw = wave >> 2, half = lane >> 4, r = lane & 15;
    const int mBase = blockIdx.x * M_TILE;
    const int bIdx = mBase / HWC, hwBase = mBase - bIdx * HWC;
    const float* fb = feature + (size_t)bIdx * CIN * HWC;

    for (int i = 0; i < 9; ++i) {
        int e = t + i * 256, p = e / GK, gk = e - p * GK;
        int g = gk / KKC, k = gk - g * KKC;
        int hwp = hwBase + p, yy = hwp / Wdim, xx = hwp - yy * Wdim;
        const float* pr = pred_shape + ((size_t)bIdx * HWC + hwp) * 4;
        const float* wo = w_offset + gk * 8;
        float offY = pr[0]*wo[0] + pr[1]*wo[1] + pr[2]*wo[2] + pr[3]*wo[3];
        float offX = pr[0]*wo[4] + pr[1]*wo[5] + pr[2]*wo[6] + pr[3]*wo[7];
        int ky = k / KS, kx = k - ky * KS;
        float py = (float)(yy - PADC + ky) + offY;
        float px = (float)(xx - PADC + kx) + offX;
        float y0f = floorf(py), x0f = floorf(px);
        int y0 = (int)y0f, x0 = (int)x0f;
        float ly = py - y0f, lx = px - x0f;
        bool yin0 = (unsigned)y0 < (unsigned)Hdim, yin1 = (unsigned)(y0+1) < (unsigned)Hdim;
        bool xin0 = (unsigned)x0 < (unsigned)Wdim, xin1 = (unsigned)(x0+1) < (unsigned)Wdim;
        float4 wv;
        wv.x = (yin0&&xin0) ? (1.f-ly)*(1.f-lx) : 0.f;
        wv.y = (yin0&&xin1) ? (1.f-ly)*lx : 0.f;
        wv.z = (yin1&&xin0) ? ly*(1.f-lx) : 0.f;
        wv.w = (yin1&&xin1) ? ly*lx : 0.f;
        int cy0 = min(max(y0,0),Hdim-1), cy1 = min(max(y0+1,0),Hdim-1);
        int cx0 = min(max(x0,0),Wdim-1), cx1 = min(max(x0+1,0),Wdim-1);
        int2 ip;
        ip.x = (cy0*Wdim+cx0) | ((cy0*Wdim+cx1) << 16);
        ip.y = (cy1*Wdim+cx0) | ((cy1*Wdim+cx1) << 16);
        *(float4*)&wtab[p*WROW + gk*4] = wv;
        *(int2*)&itab[p*IROW + gk*2]   = ip;
    }
    __syncthreads();

    v8f acc[8];
    #pragma unroll
    for (int nt = 0; nt < 8; ++nt) acc[nt] = (v8f)0.0f;
    const int prow_w = (mw*16 + r) * WROW, prow_i = (mw*16 + r) * IROW;

    for (int s = 0; s < NSTEPS; ++s) {
        const int ck0 = s * 32;
        v2f af[8];
        #pragma unroll
        for (int kq = 0; kq < 8; ++kq) {
            float vres[2];
            #pragma unroll
            for (int e2 = 0; e2 < 2; ++e2) {
                int ck = ck0 + kq*4 + half*2 + e2;
                int c = ck / KKC, k = ck - c*KKC;
                int gk = (c >> 6)*KKC + k;
                float4 wv = *(const float4*)&wtab[prow_w + gk*4];
                int2 ip = *(const int2*)&itab[prow_i + gk*2];
                const float* fp = fb + (size_t)c * HWC;
                float v = wv.x * fp[ip.x & 0xFFFF];
                v = fmaf(wv.y, fp[(unsigned)ip.x >> 16], v);
                v = fmaf(wv.z, fp[ip.y & 0xFFFF], v);
                v = fmaf(wv.w, fp[(unsigned)ip.y >> 16], v);
                vres[e2] = v;
            }
            af[kq].x = vres[0]; af[kq].y = vres[1];
        }
        const float* wbase = w_deform + ck0 + half*2;
        #pragma unroll
        for (int kq = 0; kq < 8; ++kq) {
            #pragma unroll
            for (int nt = 0; nt < 8; ++nt) {
                int o = nw*128 + nt*16 + r;
                float2 b2 = *(const float2*)(wbase + (size_t)o*KDIM + kq*4);
                v2f bf; bf.x = b2.x; bf.y = b2.y;
                acc[nt] = __builtin_amdgcn_wmma_f32_16x16x4_f32(
                    false, af[kq], false, bf, (short)0, acc[nt], false, false);
            }
        }
    }
    #pragma unroll
    for (int nt = 0; nt < 8; ++nt) {
        int o = nw*128 + nt*16 + r;
        float* op = out + ((size_t)bIdx*COUTC + o)*HWC + hwBase + mw*16 + half*8;
        float4 lo, hi;
        lo.x = fmaxf(acc[nt][0],0.f); lo.y = fmaxf(acc[nt][1],0.f);
        lo.z = fmaxf(acc[nt][2],0.f); lo.w = fmaxf(acc[nt][3],0.f);
        hi.x = fmaxf(acc[nt][4],0.f); hi.y = fmaxf(acc[nt][5],0.f);
        hi.z = fmaxf(acc[nt][6],0.f); hi.w = fmaxf(acc[nt][7],0.f);
        *(float4*)op = lo; *(float4*)(op+4) = hi;
    }
}

extern "C" void kernel_launch(void* const* d_in, const int* in_sizes, int n_in,
                              void* d_out, int out_size, void* d_ws, size_t ws_size,
                              hipStream_t stream) {
    const float* feature    = (const float*)d_in[0];
    const float* pred_shape = (const float*)d_in[1];
    const float* w_offset   = (const float*)d_in[2];
    const float* w_deform   = (const float*)d_in[3];
    float* out = (float*)d_out;

    const int numBlocks = (BATCH * HWC) / M_TILE;            // 288
    const size_t needW  = (size_t)KDIM * COUTC * sizeof(float); // 2.36 MB

    if (ws_size >= needW && d_ws != nullptr) {
        float* wre = (float*)d_ws;
        transpose_weights<<<dim3((KDIM * COUTC) / 256), dim3(256), 0, stream>>>(
            w_deform, wre);
        deform_conv_wmma_fast<<<dim3(numBlocks), dim3(256), 0, stream>>>(
            feature, pred_shape, w_offset, wre, out);
    } else {
        deform_conv_wmma_fallback<<<dim3(numBlocks), dim3(256), 0, stream>>>(
            feature, pred_shape, w_offset, w_deform, out);
    }
}